// GraphNeuralNetwork_83511344103477
// MI455X (gfx1250) — compile-verified
//
#include <hip/hip_runtime.h>

typedef float v2f __attribute__((ext_vector_type(2)));
typedef float v8f __attribute__((ext_vector_type(8)));

#define GNN_NODES  50000
#define GNN_EDGES  800000
#define GNN_CH     256
#define GNN_LAYERS 3

// ---------------------------------------------------------------- utilities
__global__ void gnn_zero_kernel(float* __restrict__ p, int n) {
  int i = blockIdx.x * blockDim.x + threadIdx.x;
  int stride = gridDim.x * blockDim.x;
  for (; i < n; i += stride) p[i] = 0.0f;
}

__global__ void gnn_degree_kernel(const int* __restrict__ dst,
                                  float* __restrict__ deg, int nE) {
  int e = blockIdx.x * blockDim.x + threadIdx.x;
  if (e < nE) atomicAdd(&deg[dst[e]], 1.0f);
}

__global__ void gnn_invdeg_kernel(float* __restrict__ deg, int n) {
  int i = blockIdx.x * blockDim.x + threadIdx.x;
  if (i < n) deg[i] = 1.0f / fmaxf(deg[i], 1.0f);
}

// one wave32 per edge: agg[dst] += h[src] over 256 channels (8 f32 per lane)
__global__ void gnn_scatter_kernel(const float* __restrict__ h,
                                   const int* __restrict__ dst,
                                   const int* __restrict__ src,
                                   float* __restrict__ agg, int nE) {
  int gid  = blockIdx.x * blockDim.x + threadIdx.x;
  int e    = gid >> 5;
  int lane = gid & 31;
  if (e >= nE) return;
  int s = src[e];
  int d = dst[e];
  const float4* hs = (const float4*)(h + (size_t)s * GNN_CH);
  float4 v0 = hs[lane * 2 + 0];
  float4 v1 = hs[lane * 2 + 1];
  float* ad = agg + (size_t)d * GNN_CH + lane * 8;
  atomicAdd(ad + 0, v0.x); atomicAdd(ad + 1, v0.y);
  atomicAdd(ad + 2, v0.z); atomicAdd(ad + 3, v0.w);
  atomicAdd(ad + 4, v1.x); atomicAdd(ad + 5, v1.y);
  atomicAdd(ad + 6, v1.z); atomicAdd(ad + 7, v1.w);
}

// ---------------------------------------------------------------- WMMA GEMM
// C[M,256] = act( scale_rows(A) @ W  (+ A2 @ W2)  + bias )
// K = N = 256 fixed. 8 waves/block; wave = 64x64 tile; block = 128 rows x 256 cols.
// All DUAL/SCALE/RELU decisions are compile-time: the K-loop is branch-free.
template <bool DUAL, bool SCALE, bool RELU>
__global__ __launch_bounds__(256, 1) void gnn_gemm(
    const float* __restrict__ A,  const float* __restrict__ W,
    const float* __restrict__ bias,
    const float* __restrict__ A2, const float* __restrict__ W2,
    const float* __restrict__ rowscale,
    float* __restrict__ C, int M) {
  const int tid    = threadIdx.x;
  const int wave   = tid >> 5;
  const int lane   = tid & 31;
  const int lane16 = lane & 15;
  const int half   = lane >> 4;

  const int m0 = blockIdx.x * 128 + (wave >> 2) * 64;  // 64-row wave tile
  const int n0 = (wave & 3) * 64;                      // 64-col wave tile
  if (m0 >= M) return;                                 // wave-uniform

  // Per-subtile A row pointers; clamp OOB subtiles to the last valid 16 rows
  // (M % 16 == 0) so the K-loop needs no guards — only stores are guarded.
  const float* aptr[4];
  const float* a2ptr[4];
  float rs[4];
#pragma unroll
  for (int i = 0; i < 4; ++i) {
    int rbase = m0 + 16 * i;
    if (rbase >= M) rbase = M - 16;      // wave-uniform clamp
    const int row = rbase + lane16;
    aptr[i] = A + (size_t)row * GNN_CH;
    if (DUAL)  a2ptr[i] = A2 + (size_t)row * GNN_CH;
    if (SCALE) rs[i] = rowscale[row];
  }
  const float* wptr  = W + lane16;
  const float* w2ptr = DUAL ? (W2 + lane16) : nullptr;

  v8f acc[4][4];
#pragma unroll
  for (int i = 0; i < 4; ++i)
#pragma unroll
    for (int j = 0; j < 4; ++j) acc[i][j] = (v8f)0.0f;

#pragma unroll 4
  for (int k0 = 0; k0 < GNN_CH; k0 += 4) {
    const int kb = k0 + 2 * half;  // 8B-aligned A offset

    // ---- A fragments: one b64 load each
    v2f a[4], a2[4];
#pragma unroll
    for (int i = 0; i < 4; ++i) {
      a[i] = *(const v2f*)(aptr[i] + kb);
      if (SCALE) { a[i].x *= rs[i]; a[i].y *= rs[i]; }
      if (DUAL)  a2[i] = *(const v2f*)(a2ptr[i] + kb);
    }

    // ---- B fragments: rows kb and kb+1 of W (stride 256 -> two b32 loads)
    v2f b[4], b2[4];
#pragma unroll
    for (int j = 0; j < 4; ++j) {
      const int coff = n0 + 16 * j;
      b[j].x = wptr[(size_t)kb * GNN_CH + coff];
      b[j].y = wptr[(size_t)(kb + 1) * GNN_CH + coff];
      if (DUAL) {
        b2[j].x = w2ptr[(size_t)kb * GNN_CH + coff];
        b2[j].y = w2ptr[(size_t)(kb + 1) * GNN_CH + coff];
      }
    }

    // ---- 16 (or 32) WMMAs per K-step, straight-line
#pragma unroll
    for (int i = 0; i < 4; ++i)
#pragma unroll
      for (int j = 0; j < 4; ++j) {
        acc[i][j] = __builtin_amdgcn_wmma_f32_16x16x4_f32(
            false, a[i], false, b[j], (short)0, acc[i][j], false, false);
        if (DUAL)
          acc[i][j] = __builtin_amdgcn_wmma_f32_16x16x4_f32(
              false, a2[i], false, b2[j], (short)0, acc[i][j], false, false);
      }
  }

  // ---- epilogue: bias (+ReLU), store (row-guarded at subtile granularity)
#pragma unroll
  for (int i = 0; i < 4; ++i) {
    const int rbase = m0 + 16 * i;
    if (rbase >= M) break;  // wave-uniform
#pragma unroll
    for (int j = 0; j < 4; ++j) {
      const int col = n0 + 16 * j + lane16;
      const float bv = bias[col];
#pragma unroll
      for (int v = 0; v < 8; ++v) {
        const int row = rbase + v + 8 * half;
        float val = acc[i][j][v] + bv;
        if (RELU) val = fmaxf(val, 0.0f);
        C[(size_t)row * GNN_CH + col] = val;
      }
    }
  }
}

// ---------------------------------------------------------------- launcher
extern "C" void kernel_launch(void* const* d_in, const int* in_sizes, int n_in,
                              void* d_out, int out_size, void* d_ws, size_t ws_size,
                              hipStream_t stream) {
  (void)in_sizes; (void)n_in; (void)out_size; (void)ws_size;

  const float* x      = (const float*)d_in[0];
  const float* W_in   = (const float*)d_in[1];
  const float* b_in   = (const float*)d_in[2];
  const float* W_agg  = (const float*)d_in[3];
  const float* W_self = (const float*)d_in[4];
  const float* b_hid  = (const float*)d_in[5];
  const float* W_out  = (const float*)d_in[6];
  const float* b_out  = (const float*)d_in[7];
  const int*   eidx   = (const int*)d_in[8];
  const int* dst = eidx;               // edge_index[0]
  const int* src = eidx + GNN_EDGES;   // edge_index[1]
  float* out = (float*)d_out;

  const size_t feat = (size_t)GNN_NODES * GNN_CH;
  float* ws   = (float*)d_ws;
  float* hA   = ws;
  float* hB   = ws + feat;
  float* agg  = ws + 2 * feat;
  float* deg  = ws + 3 * feat;

  // in-degree -> inverse degree
  gnn_zero_kernel<<<256, 256, 0, stream>>>(deg, GNN_NODES);
  gnn_degree_kernel<<<(GNN_EDGES + 255) / 256, 256, 0, stream>>>(dst, deg, GNN_EDGES);
  gnn_invdeg_kernel<<<(GNN_NODES + 255) / 256, 256, 0, stream>>>(deg, GNN_NODES);

  const int gemmBlocks = (GNN_NODES + 127) / 128;

  // h = x @ W_in + b_in
  gnn_gemm<false, false, false><<<gemmBlocks, 256, 0, stream>>>(
      x, W_in, b_in, nullptr, nullptr, nullptr, hA, GNN_NODES);

  float* hcur = hA;
  float* hnext = hB;
  for (int l = 0; l < GNN_LAYERS; ++l) {
    gnn_zero_kernel<<<2048, 256, 0, stream>>>(agg, (int)feat);
    gnn_scatter_kernel<<<(GNN_EDGES * 32 + 255) / 256, 256, 0, stream>>>(
        hcur, dst, src, agg, GNN_EDGES);
    // h = relu( (agg * inv_deg) @ W_agg[l] + h @ W_self[l] + b_hid[l] )
    gnn_gemm<true, true, true><<<gemmBlocks, 256, 0, stream>>>(
        agg, W_agg + (size_t)l * GNN_CH * GNN_CH, b_hid + (size_t)l * GNN_CH,
        hcur, W_self + (size_t)l * GNN_CH * GNN_CH, deg,
        hnext, GNN_NODES);
    float* t = hcur; hcur = hnext; hnext = t;
  }

  // out = h @ W_out + b_out
  gnn_gemm<false, false, false><<<gemmBlocks, 256, 0, stream>>>(
      hcur, W_out, b_out, nullptr, nullptr, nullptr, out, GNN_NODES);
}